// Yolo_loss_11751030522479
// MI455X (gfx1250) — compile-verified
//
#include <hip/hip_runtime.h>
#include <hip/hip_bf16.h>

#define NC 80
#define NA 3
#define NL 60
#define NB 16
#define TREC 16   // floats per target record
#define PREC 16   // floats per partial record

typedef __attribute__((ext_vector_type(2))) float v2f;
typedef __attribute__((ext_vector_type(8))) float v8f;
typedef __attribute__((ext_vector_type(4))) int v4i;

typedef __attribute__((address_space(1))) v4i* g4ptr_t;  // global, 16B vector
typedef __attribute__((address_space(3))) v4i* l4ptr_t;  // LDS, 16B vector

__device__ const float ANCHW[9] = {12.f,19.f,40.f,36.f,76.f,72.f,142.f,192.f,459.f};
__device__ const float ANCHH[9] = {16.f,36.f,28.f,75.f,55.f,146.f,110.f,243.f,401.f};

__global__ void zero_ws(float* __restrict__ p, int n) {
    int i = blockIdx.x * 256 + threadIdx.x;
    if (i < n) p[i] = 0.f;
}

// One record per (scale, batch, label): box corners, anchor assignment, scatter values.
__global__ __launch_bounds__(64) void prep_targets(const float* __restrict__ labels,
                                                   float* __restrict__ tgt) {
    int s = blockIdx.x / NB;
    int b = blockIdx.x % NB;
    int l = threadIdx.x;
    if (l >= NL) return;
    float stride = (float)(8 << s);
    int f = 608 >> (3 + s);
    const float* lab = labels + ((size_t)b * NL + l) * 5;
    float x1 = lab[0], y1 = lab[1], x2 = lab[2], y2 = lab[3], cls = lab[4];
    float valid = (x1 + y1 + x2 + y2 + cls) > 0.f ? 1.f : 0.f;
    float tx = (x1 + x2) / (2.f * stride);
    float ty = (y1 + y2) / (2.f * stride);
    float tw = (x2 - x1) / stride;
    float th = (y2 - y1) / stride;
    int ii = (int)tx;
    int jj = (int)ty;
    // best anchor over all 9 (scaled to this stride); first-max wins like jnp.argmax
    int best = 0; float bestv = -1.f;
    for (int k = 0; k < 9; k++) {
        float rw = ANCHW[k] / stride, rh = ANCHH[k] / stride;
        float mw = fminf(tw, rw), mh = fminf(th, rh);
        float inter = (mw > 0.f && mh > 0.f) ? mw * mh : 0.f;
        float aiou = inter / (tw * th + rw * rh - inter);
        if (aiou > bestv) { bestv = aiou; best = k; }
    }
    int bn = best % 3;
    bool use = (valid > 0.f) && (best / 3 == s) &&
               ii >= 0 && ii < f && jj >= 0 && jj < f;   // scatter drop semantics
    float maw = ANCHW[3 * s + bn] / stride, mah = ANCHH[3 * s + bn] / stride;
    float* r = tgt + (((size_t)s * NB + b) * NL + l) * TREC;
    r[0]  = tx - tw * 0.5f;                 // x1 (grid units)
    r[1]  = ty - th * 0.5f;                 // y1
    r[2]  = tx + tw * 0.5f;                 // x2
    r[3]  = ty + th * 0.5f;                 // y2
    r[4]  = tw * th;                        // target area
    r[5]  = valid;                          // valid mask for iou
    r[6]  = use ? (float)bn : -1.f;         // assigned anchor (or -1)
    r[7]  = (float)jj;                      // cell y
    r[8]  = (float)ii;                      // cell x
    r[9]  = tx - (float)ii;                 // t0
    r[10] = ty - (float)jj;                 // t1
    r[11] = logf(tw / maw + 1e-16f);        // t2
    r[12] = logf(th / mah + 1e-16f);        // t3
    r[13] = sqrtf(2.f - tw * th / ((float)f * (float)f)); // sc
    r[14] = cls;
    r[15] = 0.f;
}

__device__ __forceinline__ float sigm(float x) { return 1.f / (1.f + __expf(-x)); }

// One thread per (b, a, y, x) cell; blockIdx.y = batch so the LDS target table is uniform.
__global__ __launch_bounds__(256) void yolo_main(const float* __restrict__ raw,
                                                 const float* __restrict__ tgt,
                                                 float* __restrict__ partials,
                                                 int s, int f, int recBase) {
    __shared__ float sh[NL * TREC];
    __shared__ float wsum[8][5];
    int b = blockIdx.y;
    int tid = threadIdx.x;
    const float* tb = tgt + ((size_t)s * NB + b) * NL * TREC;

    // Stage the 60x16-float target table into LDS via the CDNA5 async-to-LDS path
    // (ASYNCcnt-tracked, bypasses VGPRs). AS3 pointers are the low 32 bits of the
    // generic LDS address (ISA 10.2 aperture rules), AS1 is value-identical to
    // generic for global, so integer casts are exact.
#if __has_builtin(__builtin_amdgcn_global_load_async_to_lds_b128) && \
    __has_builtin(__builtin_amdgcn_s_wait_asynccnt)
    if (tid < (NL * TREC) / 4) {
        const float* g = tb + tid * 4;
        float* l = sh + tid * 4;
        __builtin_amdgcn_global_load_async_to_lds_b128(
            (g4ptr_t)(uintptr_t)g,
            (l4ptr_t)(uint32_t)(uintptr_t)l,
            0, 0);
    }
    __builtin_amdgcn_s_wait_asynccnt(0);
    __syncthreads();
#else
    for (int i = tid; i < NL * TREC; i += 256) sh[i] = tb[i];
    __syncthreads();
#endif

    int id = blockIdx.x * 256 + tid;
    int cells = 3 * f * f;
    float lxy = 0.f, lwh = 0.f, lob = 0.f, lcl = 0.f, l2 = 0.f;
    if (id < cells) {
        int x = id % f;
        int y = (id / f) % f;
        int a = id / (f * f);
        float stride = (float)(8 << s);
        float maw = ANCHW[3 * s + a] / stride, mah = ANCHH[3 * s + a] / stride;
        size_t cs = (size_t)f * f;
        const float* base = raw + (((size_t)b * (NA * (5 + NC)) + a * (5 + NC)) * f + y) * (size_t)f + x;
        // only channels 0..4 needed for every cell (coalesced along x)
        float c0 = base[0], c1 = base[cs], c2 = base[2 * cs], c3 = base[3 * cs], c4 = base[4 * cs];
        float sx = sigm(c0), sy = sigm(c1), obj = sigm(c4);
        float pw = __expf(c2) * maw, ph = __expf(c3) * mah;
        float px = sx + (float)x, py = sy + (float)y;
        float p_l = px - pw * 0.5f, p_t = py - ph * 0.5f;
        float p_r = px + pw * 0.5f, p_b = py + ph * 0.5f;
        float ap = pw * ph;
        float maxiou = 0.f;
        int lm = -1;
        float fa = (float)a, fy = (float)y, fx = (float)x;
        for (int l = 0; l < NL; l++) {
            const float* r = sh + l * TREC;
            float tlx = fmaxf(p_l, r[0]);
            float tly = fmaxf(p_t, r[1]);
            float brx = fminf(p_r, r[2]);
            float bry = fminf(p_b, r[3]);
            float iw = brx - tlx, ih = bry - tly;
            float ai = (iw > 0.f && ih > 0.f) ? iw * ih : 0.f;
            float iou = ai / (ap + r[4] - ai) * r[5];
            maxiou = fmaxf(maxiou, iou);
            if (r[6] == fa && r[7] == fy && r[8] == fx) lm = l;  // last match wins
        }
        if (lm >= 0) {
            // target cell: obj_mask forced to 1, m = 1
            const float* r = sh + lm * TREC;
            float t0 = r[9], t1 = r[10], t2 = r[11], t3 = r[12], sc = r[13];
            int ci = (int)r[14];
            float sc2 = sc * sc;
            lob += -__logf(fmaxf(obj, 1e-12f));
            l2  += (obj - 1.f) * (obj - 1.f);
            lxy += sc2 * (-(t0 * __logf(fmaxf(sx, 1e-12f)) + (1.f - t0) * log1pf(-fminf(sx, 1.f - 1e-7f)))
                          - (t1 * __logf(fmaxf(sy, 1e-12f)) + (1.f - t1) * log1pf(-fminf(sy, 1.f - 1e-7f))));
            l2  += (sx - t0) * (sx - t0) + (sy - t1) * (sy - t1);
            float dw = c2 - t2, dh = c3 - t3;
            lwh += 0.5f * sc2 * (dw * dw + dh * dh);
            l2  += sc2 * (dw * dw + dh * dh);
            for (int c = 0; c < NC; c++) {
                float pc = sigm(base[(size_t)(5 + c) * cs]);
                float tc = (c == ci) ? 1.f : 0.f;
                lcl += -(tc * __logf(fmaxf(pc, 1e-12f)) + (1.f - tc) * log1pf(-fminf(pc, 1.f - 1e-7f)));
                float d = pc - tc;
                l2 += d * d;
            }
        } else if (maxiou <= 0.5f) {
            lob += -log1pf(-fminf(obj, 1.f - 1e-7f));
            l2  += obj * obj;
        }
    }

    // wave32 butterfly + cross-wave LDS reduction
    float v[5] = {lxy, lwh, lob, lcl, l2};
    for (int c = 0; c < 5; c++) {
        float t = v[c];
        for (int off = 16; off > 0; off >>= 1) t += __shfl_xor(t, off, 32);
        if ((tid & 31) == 0) wsum[tid >> 5][c] = t;
    }
    __syncthreads();
    if (tid == 0) {
        int rec = recBase + blockIdx.y * gridDim.x + blockIdx.x;
        float* p = partials + (size_t)rec * PREC;
        for (int c = 0; c < 5; c++) {
            float acc = 0.f;
            for (int w = 0; w < 8; w++) acc += wsum[w][c];
            p[c] = acc;  // components 5..15 stay zero (zero_ws)
        }
    }
}

// Exact f32 reduction via V_WMMA_F32_16X16X4_F32: D = ones(16x4) x B(4x16) + C
// sums 4 records per issue per column; result invariant to K-slot assignment.
__global__ __launch_bounds__(32) void final_reduce(const float* __restrict__ partials,
                                                   float* __restrict__ out, int nrec) {
    int lane = threadIdx.x;
    int col = lane & 15;
    int hi = lane >> 4;
    v2f ones; ones[0] = 1.f; ones[1] = 1.f;
    v8f acc = {};
    for (int r = 0; r < nrec; r += 4) {
        v2f bv;
        bv[0] = partials[(size_t)(r + (hi ? 2 : 0)) * PREC + col];
        bv[1] = partials[(size_t)(r + (hi ? 3 : 1)) * PREC + col];
        acc = __builtin_amdgcn_wmma_f32_16x16x4_f32(false, ones, false, bv,
                                                    (short)0, acc, false, false);
    }
    __shared__ float sums[16];
    if (lane < 16) sums[lane] = acc[0];   // D row 0: N = lane
    __syncthreads();
    if (lane == 0) {
        float a = sums[0], w = sums[1], o = sums[2], c = sums[3], q = sums[4];
        out[0] = a + w + o + c;
        out[1] = a; out[2] = w; out[3] = o; out[4] = c; out[5] = q;
    }
}

extern "C" void kernel_launch(void* const* d_in, const int* in_sizes, int n_in,
                              void* d_out, int out_size, void* d_ws, size_t ws_size,
                              hipStream_t stream) {
    const float* xin0   = (const float*)d_in[0];
    const float* xin1   = (const float*)d_in[1];
    const float* xin2   = (const float*)d_in[2];
    const float* labels = (const float*)d_in[3];
    float* out = (float*)d_out;
    float* ws  = (float*)d_ws;

    float* tgt      = ws;                       // 3*16*60*16 = 46080 floats (184 KB)
    float* partials = ws + 3 * NB * NL * TREC;  // block partial records

    const int f0 = 76, f1 = 38, f2 = 19;
    const int g0 = (3 * f0 * f0 + 255) / 256;   // 68
    const int g1 = (3 * f1 * f1 + 255) / 256;   // 17
    const int g2 = (3 * f2 * f2 + 255) / 256;   // 5
    const int nrec  = (g0 + g1 + g2) * NB;      // 1440
    const int nrec4 = (nrec + 3) & ~3;
    const int zn = nrec4 * PREC;

    zero_ws<<<(zn + 255) / 256, 256, 0, stream>>>(partials, zn);
    prep_targets<<<3 * NB, 64, 0, stream>>>(labels, tgt);
    yolo_main<<<dim3(g0, NB), 256, 0, stream>>>(xin0, tgt, partials, 0, f0, 0);
    yolo_main<<<dim3(g1, NB), 256, 0, stream>>>(xin1, tgt, partials, 1, f1, g0 * NB);
    yolo_main<<<dim3(g2, NB), 256, 0, stream>>>(xin2, tgt, partials, 2, f2, (g0 + g1) * NB);
    final_reduce<<<1, 32, 0, stream>>>(partials, out, nrec4);
}